// Net_188978561332
// MI455X (gfx1250) — compile-verified
//
#include <hip/hip_runtime.h>
#include <math.h>

typedef float v2f __attribute__((ext_vector_type(2)));
typedef float v8f __attribute__((ext_vector_type(8)));
typedef int   v4i __attribute__((vector_size(4 * sizeof(int))));  // matches builtin pointee

#define NCOLS        10
#define TILE_ROWS    512
#define TILE_FLOATS  (TILE_ROWS * NCOLS)   // 5120 floats = 20 KB per buffer
#define TILE_F4      (TILE_FLOATS / 4)     // 1280
#define BLOCK_THREADS 256
#define F4_PER_THREAD (TILE_F4 / BLOCK_THREADS)  // 5
#define WAVES_PER_BLOCK 8
#define ROWS_PER_WAVE (TILE_ROWS / WAVES_PER_BLOCK) // 64

// ---- CDNA5 async memory->LDS path (guarded: degrades to sync loads) -------
#if defined(__has_builtin)
#  if __has_builtin(__builtin_amdgcn_global_load_async_to_lds_b128)
#    define USE_ASYNC_LDS 1
#  endif
#endif
#ifndef USE_ASYNC_LDS
#  define USE_ASYNC_LDS 0
#endif

#define AS1 __attribute__((address_space(1)))
#define AS3 __attribute__((address_space(3)))

#if USE_ASYNC_LDS
template <int N>
__device__ __forceinline__ void s_wait_async() {
#if __has_builtin(__builtin_amdgcn_s_wait_asynccnt)
    __builtin_amdgcn_s_wait_asynccnt((unsigned short)N);
#else
    asm volatile("s_wait_asynccnt %0" :: "i"(N) : "memory");
#endif
}
#endif

// ---------------------------------------------------------------------------
// Stage 1: Gram matrix G = X^T X (10x10 padded to 16x16) via
// V_WMMA_F32_16X16X4_F32, double-buffered 512-row tiles streamed into LDS
// (async-to-LDS DMA when available; clean unguarded b128 path otherwise,
// with the zero-padded tail tile handled separately).
// ---------------------------------------------------------------------------
__global__ __launch_bounds__(BLOCK_THREADS)
void gram_wmma_kernel(const float* __restrict__ X,
                      float* __restrict__ partial,   // [gridDim.x * 256]
                      long long totalRows)
{
    __shared__ __align__(16) float lds[2 * TILE_FLOATS];   // 40 KB

    const int tid   = threadIdx.x;
    const int wave  = tid >> 5;
    const int lane  = tid & 31;
    const int col   = lane & 15;        // M (and N) index 0..15
    const int khalf = lane >> 4;        // 0: K=0,1   1: K=2,3
    const bool colOK = (col < NCOLS);

    const long long totalF4  = (totalRows * NCOLS) >> 2;   // 10M*10 % 4 == 0
    const long long numTiles = (totalRows + TILE_ROWS - 1) / TILE_ROWS;
    const long long stride   = gridDim.x;

    v8f c = {0.f, 0.f, 0.f, 0.f, 0.f, 0.f, 0.f, 0.f};
    const float4* X4 = reinterpret_cast<const float4*>(X);

    // Issue (async) or copy (sync) one tile into LDS buffer `buf`.
    // Returns true if async loads were issued for this tile.
    auto load_tile = [&](int buf, long long tile) -> bool {
        const long long base4 = tile * (long long)TILE_F4;
        float4* dst = reinterpret_cast<float4*>(lds + (size_t)buf * TILE_FLOATS);
        if (base4 + TILE_F4 <= totalF4) {
#if USE_ASYNC_LDS
            const float4* g = X4 + base4 + tid;
            float4* l = dst + tid;
            #pragma unroll
            for (int j = 0; j < F4_PER_THREAD; ++j)
                __builtin_amdgcn_global_load_async_to_lds_b128(
                    (AS1 v4i*)(g + j * BLOCK_THREADS),
                    (AS3 v4i*)(l + j * BLOCK_THREADS), 0, 0);
            return true;
#else
            #pragma unroll
            for (int j = 0; j < F4_PER_THREAD; ++j) {
                const int s = tid + j * BLOCK_THREADS;
                dst[s] = X4[base4 + s];
            }
            return false;
#endif
        } else {
            // tail tile: guarded loads, zero padding (pads contribute 0 to Gram)
            #pragma unroll
            for (int j = 0; j < F4_PER_THREAD; ++j) {
                const int s = tid + j * BLOCK_THREADS;
                const long long g = base4 + s;
                float4 v = make_float4(0.f, 0.f, 0.f, 0.f);
                if (g < totalF4) v = X4[g];
                dst[s] = v;
            }
            return false;
        }
    };

    int cur = 0;
    const long long t0 = blockIdx.x;
    if (t0 < numTiles) load_tile(0, t0);

    for (long long t = t0; t < numTiles; t += stride) {
        const long long nt = t + stride;
        bool nextAsync = false;
        if (nt < numTiles) nextAsync = load_tile(cur ^ 1, nt);
        (void)nextAsync;

#if USE_ASYNC_LDS
        // Async loads complete in order: <= F4_PER_THREAD outstanding means the
        // current tile's loads have all landed in LDS.
        if (nextAsync) s_wait_async<F4_PER_THREAD>();
        else           s_wait_async<0>();
#else
        const long long pf = t + 2 * stride;
        if (pf < numTiles) {
            const char* p = (const char*)(X4 + pf * TILE_F4) + (size_t)tid * 64;
            __builtin_prefetch(p, 0, 1);
        }
#endif
        __syncthreads();

        // A fragment layout (16x4 f32): lane = M + 16*(k/2); VGPR0:K={0,2},
        // VGPR1:K={1,3}. A[m][k] = X[r0+k][m]; the 4x16 B fragment of X is
        // lane-identical, so D += A*B accumulates G[m][n] = sum_k X[k][m]X[k][n].
        const float* tl = lds + (size_t)cur * TILE_FLOATS;
        const int rbase = wave * ROWS_PER_WAVE;
        #pragma unroll
        for (int kc = 0; kc < ROWS_PER_WAVE / 4; ++kc) {
            const int r0 = rbase + kc * 4 + khalf * 2;
            v2f a;
            a.x = colOK ? tl[r0 * NCOLS + col]       : 0.0f;
            a.y = colOK ? tl[(r0 + 1) * NCOLS + col] : 0.0f;
            c = __builtin_amdgcn_wmma_f32_16x16x4_f32(
                    false, a, false, a, (short)0, c, false, false);
        }
        __syncthreads();   // everyone done reading `cur` before it is refilled
        cur ^= 1;
    }

    // C/D layout: VGPR v, lanes0-15 -> M=v,N=lane ; lanes16-31 -> M=v+8,N=lane-16
    const int M0 = khalf * 8;
    #pragma unroll
    for (int v = 0; v < 8; ++v)
        lds[wave * 256 + (M0 + v) * 16 + col] = c[v];
    __syncthreads();

    // per-block reduction across the 8 waves, one 16x16 partial per block
    float s = 0.f;
    #pragma unroll
    for (int w = 0; w < WAVES_PER_BLOCK; ++w)
        s += lds[w * 256 + tid];
    partial[(size_t)blockIdx.x * 256 + tid] = s;
}

// ---------------------------------------------------------------------------
// Stage 2: deterministic reduction of per-block partial Grams
// ---------------------------------------------------------------------------
__global__ __launch_bounds__(256)
void reduce_partial_kernel(const float* __restrict__ partial,
                           float* __restrict__ G, int nb)
{
    const int t = threadIdx.x;  // 256 threads, one block
    float s = 0.f;
    for (int b = 0; b < nb; ++b) s += partial[(size_t)b * 256 + t];
    G[t] = s;
}

// ---------------------------------------------------------------------------
// Stage 3: tiny scalar post-processing (10x10 distances, top-k, entropy, xi)
// Writes dense per-column weights wcol[10] (avoids dynamic indexing later).
// ---------------------------------------------------------------------------
__global__ void select_kernel(const float* __restrict__ G,
                              const float* __restrict__ r_all,
                              float* __restrict__ wcol)
{
    if (threadIdx.x != 0 || blockIdx.x != 0) return;
    const int n = NCOLS;
    const float LOG4 = 1.3862943611198906f;

    float cd[NCOLS][NCOLS];
    for (int i = 0; i < n; ++i)
        for (int j = 0; j < n; ++j) {
            float d2 = G[i * 16 + i] + G[j * 16 + j] - 2.0f * G[i * 16 + j];
            d2 = d2 > 0.f ? d2 : 0.f;
            cd[i][j] = d2 > 0.f ? sqrtf(d2) : 0.f;
        }

    // per-row sum of the 4 smallest distances (includes self = 0)
    float rs[NCOLS];
    for (int i = 0; i < n; ++i) {
        float v[NCOLS];
        for (int j = 0; j < n; ++j) v[j] = cd[i][j];
        for (int a = 0; a < 4; ++a) {
            int mi = a;
            for (int b = a + 1; b < n; ++b) if (v[b] < v[mi]) mi = b;
            float t = v[a]; v[a] = v[mi]; v[mi] = t;
        }
        rs[i] = v[0] + v[1] + v[2] + v[3];
    }

    // idx = 4 rows with smallest sums (ties -> lower index, matching top_k)
    int idx[4];
    bool used[NCOLS] = {false};
    for (int a = 0; a < 4; ++a) {
        int mi = -1;
        for (int b = 0; b < n; ++b) {
            if (used[b]) continue;
            if (mi < 0 || rs[b] < rs[mi]) mi = b;
        }
        used[mi] = true; idx[a] = mi;
    }

    float score[4];
    for (int t = 0; t < 4; ++t) score[t] = rs[idx[t]];

    // Es = entropy of min-normalized (inverted) score
    float Es;
    {
        float mn = score[0], mx = score[0];
        for (int t = 1; t < 4; ++t) { mn = fminf(mn, score[t]); mx = fmaxf(mx, score[t]); }
        const float denom = mx - mn;
        const float safe = (denom != 0.f) ? denom : 1.f;
        float nrm[4], S = 0.f;
        for (int t = 0; t < 4; ++t) {
            nrm[t] = (denom != 0.f) ? ((mx - score[t]) / safe) : score[t];
            S += nrm[t];
        }
        float ent = 0.f;
        for (int t = 0; t < 4; ++t) { float q = nrm[t] / S; if (q > 0.f) ent += q * logf(q); }
        Es = -ent / LOG4;
    }

    // reputation bump + Er
    float rep[4];
    for (int t = 0; t < 4; ++t) {
        float r = r_all[idx[t]];
        rep[t] = r + ((r < 1.f) ? 0.05f : 0.f);
    }
    float Er;
    {
        float mn = rep[0], mx = rep[0];
        for (int t = 1; t < 4; ++t) { mn = fminf(mn, rep[t]); mx = fmaxf(mx, rep[t]); }
        const float denom = mx - mn;
        const float safe = (denom != 0.f) ? denom : 1.f;
        float nrm[4], S = 0.f;
        for (int t = 0; t < 4; ++t) {
            nrm[t] = (denom != 0.f) ? ((rep[t] - mn) / safe) : rep[t];
            S += nrm[t];
        }
        float ent = 0.f;
        for (int t = 0; t < 4; ++t) { float q = nrm[t] / S; if (q > 0.f) ent += q * logf(q); }
        Er = -ent / LOG4;
    }

    const float den   = 2.f - Es - Er;
    const float alpha = (1.f - Es) / den;
    const float beta  = (1.f - Er) / den;
    float w[4], Sw = 0.f;
    for (int t = 0; t < 4; ++t) { w[t] = alpha * score[t] + beta * rep[t]; Sw += w[t]; }

    for (int j = 0; j < NCOLS; ++j) {
        float wc = 0.f;
        for (int t = 0; t < 4; ++t) if (idx[t] == j) wc += w[t] / Sw;
        wcol[j] = wc;
    }
}

// ---------------------------------------------------------------------------
// Stage 4: out[r] = dot(row_r, wcol). 2 rows / thread: 5x b128 loads, 1x b64 store.
// ---------------------------------------------------------------------------
__global__ __launch_bounds__(256)
void weighted_sum_kernel(const float* __restrict__ X,
                         const float* __restrict__ wcol,
                         float* __restrict__ out,
                         long long totalRows)
{
    float w[NCOLS];
    #pragma unroll
    for (int j = 0; j < NCOLS; ++j) w[j] = wcol[j];

    const long long pairs  = totalRows >> 1;
    const long long stride = (long long)gridDim.x * blockDim.x;
    const float4* X4 = reinterpret_cast<const float4*>(X);

    for (long long p = (long long)blockIdx.x * blockDim.x + threadIdx.x;
         p < pairs; p += stride) {
        const long long b4 = p * 5;  // 2 rows = 20 floats = 5 float4, 16B aligned
        float4 v0 = X4[b4 + 0];
        float4 v1 = X4[b4 + 1];
        float4 v2 = X4[b4 + 2];
        float4 v3 = X4[b4 + 3];
        float4 v4 = X4[b4 + 4];
        float o0 = v0.x*w[0] + v0.y*w[1] + v0.z*w[2] + v0.w*w[3]
                 + v1.x*w[4] + v1.y*w[5] + v1.z*w[6] + v1.w*w[7]
                 + v2.x*w[8] + v2.y*w[9];
        float o1 = v2.z*w[0] + v2.w*w[1]
                 + v3.x*w[2] + v3.y*w[3] + v3.z*w[4] + v3.w*w[5]
                 + v4.x*w[6] + v4.y*w[7] + v4.z*w[8] + v4.w*w[9];
        reinterpret_cast<float2*>(out)[p] = make_float2(o0, o1);
    }

    if (blockIdx.x == 0 && threadIdx.x == 0 && (totalRows & 1)) {
        const long long r = totalRows - 1;
        float o = 0.f;
        for (int j = 0; j < NCOLS; ++j) o += X[r * NCOLS + j] * w[j];
        out[r] = o;
    }
}

extern "C" void kernel_launch(void* const* d_in, const int* in_sizes, int n_in,
                              void* d_out, int out_size, void* d_ws, size_t ws_size,
                              hipStream_t stream)
{
    const float* X     = (const float*)d_in[0];
    const float* r_all = (const float*)d_in[1];
    float* out = (float*)d_out;
    const long long totalRows = (long long)in_sizes[0] / NCOLS;

    float* ws = (float*)d_ws;
    int NB = 512;
    size_t need = ((size_t)NB * 256 + 256 + 16) * sizeof(float);
    while (need > ws_size && NB > 1) {
        NB >>= 1;
        need = ((size_t)NB * 256 + 256 + 16) * sizeof(float);
    }
    float* partial = ws;
    float* G       = ws + (size_t)NB * 256;
    float* wcol    = G + 256;

    gram_wmma_kernel<<<NB, BLOCK_THREADS, 0, stream>>>(X, partial, totalRows);
    reduce_partial_kernel<<<1, 256, 0, stream>>>(partial, G, NB);
    select_kernel<<<1, 32, 0, stream>>>(G, r_all, wcol);
    weighted_sum_kernel<<<2048, 256, 0, stream>>>(X, wcol, out, totalRows);
}